// HeteroGATLayer_50568944943220
// MI455X (gfx1250) — compile-verified
//
#include <hip/hip_runtime.h>

// ---------------------------------------------------------------------------
// HeteroGAT layer for MI455X (gfx1250, wave32).
//  - Dense projections via V_WMMA_F32_16X16X4_F32 (full fp32 precision; GEMM
//    is HBM-bound so low-precision WMMA buys nothing here).
//  - Edge attention: wave32-per-edge, coalesced 256-wide gathers (hit 192MB L2
//    since z_op=102MB fits), float atomics for segment sums.
// ---------------------------------------------------------------------------

typedef __attribute__((ext_vector_type(2))) float v2f;
typedef __attribute__((ext_vector_type(8))) float v8f;

#define OUT_DIM 256
#define HEADS   4
#define DK      64
#define ATT_ROW (2 * DK + 1)   // 129
#define LEAKY   0.2f
#define EPSV    1e-10f

// ---------------------------------------------------------------------------
// Z[M x 256] = A[M x K] @ W^T[K x 256] + bias    (K = 128 or 64, M % 16 == 0)
// One wave computes one 16x16 output tile with K/4 chained f32 WMMAs.
// WMMA 16x16x4 f32 VGPR layout (ISA 7.12.2):
//   A frag : lane l holds (M = m0+(l&15), K = k + 2*(l>>4) .. +1)  -> float2
//   B frag : lane l holds (K = k + 2*(l>>4) .. +1, N = n0+(l&15))
//            B(k,n) = W[n][k]  (W row-major [256 x K])             -> float2
//   C/D    : VGPR v: M = m0 + v + 8*(l>>4), N = n0 + (l&15)
// ---------------------------------------------------------------------------
template <int K>
__global__ void __launch_bounds__(256)
gemm_wmma_f32(const float* __restrict__ A,     // M x K
              const float* __restrict__ W,     // 256 x K
              const float* __restrict__ bias,  // 256
              float* __restrict__ Z)           // M x 256
{
    const int lane  = threadIdx.x;                       // 0..31
    const int wave  = threadIdx.y;                       // 0..7
    const int m0    = blockIdx.x * 16;
    const int n0    = (blockIdx.y * 8 + wave) * 16;      // 16 N-tiles total
    const int l15   = lane & 15;
    const int khalf = (lane >> 4) * 2;                   // 0 or 2

    const float* aptr = A + (size_t)(m0 + l15) * K + khalf;
    const float* bptr = W + (size_t)(n0 + l15) * K + khalf;

    v8f c = {};
#pragma unroll
    for (int k = 0; k < K; k += 4) {
        v2f a = *(const v2f*)(aptr + k);
        v2f b = *(const v2f*)(bptr + k);
        // (neg_a, A, neg_b, B, c_mod, C, reuse_a, reuse_b)
        c = __builtin_amdgcn_wmma_f32_16x16x4_f32(false, a, false, b,
                                                  (short)0, c, false, false);
    }

    const float bn    = bias[n0 + l15];
    const int   mbase = m0 + (lane >> 4) * 8;
    float* zptr = Z + (size_t)mbase * OUT_DIM + n0 + l15;
#pragma unroll
    for (int v = 0; v < 8; ++v)
        zptr[(size_t)v * OUT_DIM] = c[v] + bn;
}

// ---------------------------------------------------------------------------
// Pass 1 per edge: s[h] = <z_s, a_s[h]> + <z_d, a_d[h]> + feat*a_f[h]
//                  alpha = exp(leakyrelu(s)); denom[dst][h] += alpha
// One wave32 per edge; lane l covers dims {32*i + l}, head = i>>1.
// ---------------------------------------------------------------------------
__global__ void __launch_bounds__(256)
edge_score(const float* __restrict__ zs, const float* __restrict__ zd,
           const int* __restrict__ src, const int* __restrict__ dst,
           const float* __restrict__ feat, const float* __restrict__ att,
           float* __restrict__ alpha, float* __restrict__ denom, int E)
{
    const int warp = threadIdx.x >> 5;
    const int lane = threadIdx.x & 31;
    const int e = blockIdx.x * 8 + warp;
    if (e >= E) return;

    const int s = src[e];
    const int d = dst[e];
    const float* zse = zs + (size_t)s * OUT_DIM;
    const float* zde = zd + (size_t)d * OUT_DIM;

    float acc[HEADS];
#pragma unroll
    for (int h = 0; h < HEADS; ++h) {
        const float* as = att + h * ATT_ROW;       // a_src[h][0..63]
        const float* ad = as + DK;                 // a_dst[h][0..63]
        float a = 0.f;
#pragma unroll
        for (int j = 0; j < 2; ++j) {
            const int dk = j * 32 + lane;          // 0..63 within head
            const int dd = h * DK + dk;            // 0..255 global dim
            a += zse[dd] * as[dk] + zde[dd] * ad[dk];
        }
#pragma unroll
        for (int off = 16; off > 0; off >>= 1)     // wave32 butterfly
            a += __shfl_xor(a, off, 32);
        acc[h] = a;
    }

    if (lane == 0) {
        const float f = feat[e];
#pragma unroll
        for (int h = 0; h < HEADS; ++h) {
            float sc = acc[h] + f * att[h * ATT_ROW + 2 * DK];
            sc = sc > 0.f ? sc : LEAKY * sc;
            float al = __expf(sc);
            alpha[(size_t)e * HEADS + h] = al;
            atomicAdd(&denom[(size_t)d * HEADS + h], al);
        }
    }
}

// ---------------------------------------------------------------------------
// Pass 2 per edge: out[dst][h][:] += z_s[h][:] * alpha/(denom[dst][h]+eps)
// ---------------------------------------------------------------------------
__global__ void __launch_bounds__(256)
edge_scatter(const float* __restrict__ zs,
             const int* __restrict__ src, const int* __restrict__ dst,
             const float* __restrict__ alpha, const float* __restrict__ denom,
             float* __restrict__ out, int E)
{
    const int warp = threadIdx.x >> 5;
    const int lane = threadIdx.x & 31;
    const int e = blockIdx.x * 8 + warp;
    if (e >= E) return;

    const int s = src[e];
    const int d = dst[e];

    float w[HEADS];
#pragma unroll
    for (int h = 0; h < HEADS; ++h)
        w[h] = alpha[(size_t)e * HEADS + h] /
               (denom[(size_t)d * HEADS + h] + EPSV);

    const float* zse = zs + (size_t)s * OUT_DIM;
    float* od = out + (size_t)d * OUT_DIM;
#pragma unroll
    for (int i = 0; i < 8; ++i) {
        const int dd = i * 32 + lane;              // head = i>>1
        atomicAdd(&od[dd], zse[dd] * w[i >> 1]);
    }
}

// ---------------------------------------------------------------------------
// res = elu(out_acc + z)   (elu: x>0 ? x : exp(x)-1)
// ---------------------------------------------------------------------------
__global__ void __launch_bounds__(256)
finalize_elu(const float* __restrict__ z, const float* __restrict__ acc,
             float* __restrict__ res, size_t n)
{
    size_t i = (size_t)blockIdx.x * blockDim.x + threadIdx.x;
    if (i >= n) return;
    float r = acc[i] + z[i];
    res[i] = r > 0.f ? r : expm1f(r);
}

// ---------------------------------------------------------------------------
extern "C" void kernel_launch(void* const* d_in, const int* in_sizes, int n_in,
                              void* d_out, int out_size, void* d_ws, size_t ws_size,
                              hipStream_t stream)
{
    const float* h_op   = (const float*)d_in[0];
    const float* h_mac  = (const float*)d_in[1];
    const float* f_seq  = (const float*)d_in[2];
    const float* f_om   = (const float*)d_in[3];
    const float* f_mo   = (const float*)d_in[4];
    const float* W_op   = (const float*)d_in[5];
    const float* b_op   = (const float*)d_in[6];
    const float* W_mac  = (const float*)d_in[7];
    const float* b_mac  = (const float*)d_in[8];
    const float* a_seq  = (const float*)d_in[9];
    const float* a_om   = (const float*)d_in[10];
    const float* a_mo   = (const float*)d_in[11];
    const int*   seq_s  = (const int*)d_in[12];
    const int*   seq_d  = (const int*)d_in[13];
    const int*   om_s   = (const int*)d_in[14];
    const int*   om_d   = (const int*)d_in[15];
    const int*   mo_s   = (const int*)d_in[16];
    const int*   mo_d   = (const int*)d_in[17];

    const int N_OP  = in_sizes[0] / 128;   // 100000 (multiple of 16)
    const int N_MAC = in_sizes[1] / 64;    // 20000  (multiple of 16)
    const int E     = in_sizes[2];         // 200000

    // Workspace layout (floats)
    float* ws = (float*)d_ws;
    float* z_op    = ws;
    float* z_mac   = z_op    + (size_t)N_OP  * OUT_DIM;
    float* out_op  = z_mac   + (size_t)N_MAC * OUT_DIM;   // zeroed region start
    float* out_mac = out_op  + (size_t)N_OP  * OUT_DIM;
    float* den_sq  = out_mac + (size_t)N_MAC * OUT_DIM;
    float* den_om  = den_sq  + (size_t)N_OP  * HEADS;
    float* den_mo  = den_om  + (size_t)N_MAC * HEADS;
    float* alpha   = den_mo  + (size_t)N_OP  * HEADS;     // zeroed region end

    const size_t zero_floats = (size_t)(alpha - out_op);
    hipMemsetAsync(out_op, 0, zero_floats * sizeof(float), stream);

    // Dense projections via f32 WMMA. block = (32,8): 8 waves, each one N-tile.
    dim3 blk(32, 8);
    gemm_wmma_f32<128><<<dim3(N_OP  / 16, 2), blk, 0, stream>>>(h_op,  W_op,  b_op,  z_op);
    gemm_wmma_f32<64> <<<dim3(N_MAC / 16, 2), blk, 0, stream>>>(h_mac, W_mac, b_mac, z_mac);

    const int eb = (E + 7) / 8;   // 8 wave32 edges per 256-thread block

    // seq: op -> op
    edge_score  <<<eb, 256, 0, stream>>>(z_op,  z_op,  seq_s, seq_d, f_seq, a_seq, alpha, den_sq, E);
    edge_scatter<<<eb, 256, 0, stream>>>(z_op,         seq_s, seq_d, alpha, den_sq, out_op, E);
    // om: op -> mac
    edge_score  <<<eb, 256, 0, stream>>>(z_op,  z_mac, om_s,  om_d,  f_om,  a_om,  alpha, den_om, E);
    edge_scatter<<<eb, 256, 0, stream>>>(z_op,         om_s,  om_d,  alpha, den_om, out_mac, E);
    // mo: mac -> op
    edge_score  <<<eb, 256, 0, stream>>>(z_mac, z_op,  mo_s,  mo_d,  f_mo,  a_mo,  alpha, den_mo, E);
    edge_scatter<<<eb, 256, 0, stream>>>(z_mac,        mo_s,  mo_d,  alpha, den_mo, out_op, E);

    // Residual + ELU into d_out (op block, then mac block — tuple order)
    const size_t n_op_el  = (size_t)N_OP  * OUT_DIM;
    const size_t n_mac_el = (size_t)N_MAC * OUT_DIM;
    float* outp = (float*)d_out;
    finalize_elu<<<(unsigned)((n_op_el  + 255) / 256), 256, 0, stream>>>(z_op,  out_op,  outp,           n_op_el);
    finalize_elu<<<(unsigned)((n_mac_el + 255) / 256), 256, 0, stream>>>(z_mac, out_mac, outp + n_op_el, n_mac_el);
}